// BGP_GNN_64725157150909
// MI455X (gfx1250) — compile-verified
//
#include <hip/hip_runtime.h>

typedef __attribute__((ext_vector_type(2))) float v2f;
typedef __attribute__((ext_vector_type(8))) float v8f;

#define HID 8

// ---------- degree / normalization ----------
__global__ void zero_f32_kernel(float* __restrict__ p, int n) {
    int i = blockIdx.x * blockDim.x + threadIdx.x;
    if (i < n) p[i] = 0.0f;
}

__global__ void deg_accum_kernel(const int* __restrict__ dst, int n_edges,
                                 float* __restrict__ deg) {
    int e = blockIdx.x * blockDim.x + threadIdx.x;
    if (e < n_edges) atomicAdd(&deg[dst[e]], 1.0f);
}

__global__ void deg_to_dis_kernel(float* __restrict__ deg_dis, int n) {
    int i = blockIdx.x * blockDim.x + threadIdx.x;
    if (i < n) deg_dis[i] = rsqrtf(deg_dis[i] + 1.0f);
}

// ---------- layer 0: in_dim == 1, h_lin = x[n] * W1 ; h_next = h_lin*dis^2 + b ----------
__global__ void layer0_transform_kernel(const float* __restrict__ x,
                                        const float* __restrict__ W1,
                                        const float* __restrict__ b1,
                                        const float* __restrict__ dis,
                                        float* __restrict__ h_lin,
                                        float* __restrict__ h_next, int n) {
    int i = blockIdx.x * blockDim.x + threadIdx.x;
    if (i >= n) return;
    float xv = x[i];
    float ds = dis[i];
    float d2 = ds * ds;
    float4 w0 = ((const float4*)W1)[0];
    float4 w1 = ((const float4*)W1)[1];
    float4 c0 = ((const float4*)b1)[0];
    float4 c1 = ((const float4*)b1)[1];
    float4 l0 = make_float4(xv * w0.x, xv * w0.y, xv * w0.z, xv * w0.w);
    float4 l1 = make_float4(xv * w1.x, xv * w1.y, xv * w1.z, xv * w1.w);
    float4* lp = (float4*)(h_lin + (size_t)i * HID);
    lp[0] = l0; lp[1] = l1;
    float4 n0 = make_float4(fmaf(l0.x, d2, c0.x), fmaf(l0.y, d2, c0.y),
                            fmaf(l0.z, d2, c0.z), fmaf(l0.w, d2, c0.w));
    float4 n1 = make_float4(fmaf(l1.x, d2, c1.x), fmaf(l1.y, d2, c1.y),
                            fmaf(l1.z, d2, c1.z), fmaf(l1.w, d2, c1.w));
    float4* np = (float4*)(h_next + (size_t)i * HID);
    np[0] = n0; np[1] = n1;
}

// ---------- layers 1-3: h_lin = relu(h_in) @ W ; h_next = h_lin*dis^2 + b ----------
// One wave = one 16-node tile. D(16x16) = A(16x4)*B(4x16)+C, two chained K=4 WMMAs.
__global__ void gcn_transform_wmma8_kernel(const float* __restrict__ h_in,
                                           const float* __restrict__ Wm,
                                           const float* __restrict__ bias,
                                           const float* __restrict__ dis,
                                           float* __restrict__ h_lin,
                                           float* __restrict__ h_next,
                                           int n_nodes) {
    const int lane = threadIdx.x & 31;
    const int wave = (blockIdx.x * blockDim.x + threadIdx.x) >> 5;
    const int m0 = wave * 16;
    if (m0 >= n_nodes) return;                 // wave-uniform exit: EXEC all-1s for WMMA

    const int half = lane >> 4;                // 0: lanes 0-15, 1: lanes 16-31
    const int lm   = lane & 15;                // M (A rows) / N (B,C,D cols)
    int node = m0 + lm;
    if (node >= n_nodes) node = n_nodes - 1;   // clamped load for tail tiles

    const float2* __restrict__ ap = (const float2*)(h_in + (size_t)node * HID);
    const int  col  = lm & 7;                  // in-bounds W column
    const bool live = lm < HID;                // only N<8 columns are real

    v8f c = {};                                // 16x16 f32 accumulator (8 VGPRs)
#pragma unroll
    for (int ck = 0; ck < 2; ++ck) {
        const int k0 = 4 * ck + 2 * half;      // lane supplies local K = 2h, 2h+1
        float2 av = ap[2 * ck + half];         // unconditional b64 load
        av.x = fmaxf(av.x, 0.0f);              // relu on input (layers 2-4)
        av.y = fmaxf(av.y, 0.0f);
        v2f a; a.x = av.x; a.y = av.y;
        float w0 = Wm[k0 * HID + col];         // in-bounds, broadcast-cached
        float w1 = Wm[(k0 + 1) * HID + col];
        v2f bm; bm.x = live ? w0 : 0.0f; bm.y = live ? w1 : 0.0f;  // branchless
        c = __builtin_amdgcn_wmma_f32_16x16x4_f32(
                false, a, false, bm, (short)0, c, false, false);
    }

    // D layout: VGPR r -> (M = r + 8*half, N = lm). Only N < 8 is live.
    if (live) {
        const float bj   = bias[col];
        const int  rbase = m0 + half * 8;
        if (rbase + 8 <= n_nodes) {            // full-tile fast path (wave-uniform-ish)
#pragma unroll
            for (int r = 0; r < 8; ++r) {
                int   row = rbase + r;
                float v   = c[r];
                float ds  = dis[row];
                h_lin [(size_t)row * HID + col] = v;
                h_next[(size_t)row * HID + col] = fmaf(v, ds * ds, bj);
            }
        } else {
#pragma unroll
            for (int r = 0; r < 8; ++r) {
                int row = rbase + r;
                if (row < n_nodes) {
                    float v  = c[r];
                    float ds = dis[row];
                    h_lin [(size_t)row * HID + col] = v;
                    h_next[(size_t)row * HID + col] = fmaf(v, ds * ds, bj);
                }
            }
        }
    }
}

// ---------- edge scatter: h_next[dst] += h_lin[src] * dis[src]*dis[dst] ----------
__global__ void gcn_scatter_kernel(const int* __restrict__ src,
                                   const int* __restrict__ dst,
                                   const float* __restrict__ dis,
                                   const float* __restrict__ h_lin,
                                   float* __restrict__ h_next, int n_edges) {
    int e = blockIdx.x * blockDim.x + threadIdx.x;
    if (e >= n_edges) return;
    int s = src[e];
    int d = dst[e];
    float w = dis[s] * dis[d];
    const float4* hl = (const float4*)(h_lin + (size_t)s * HID);
    float4 h0 = hl[0];
    float4 h1 = hl[1];
    float* out = h_next + (size_t)d * HID;
    atomicAdd(out + 0, h0.x * w);
    atomicAdd(out + 1, h0.y * w);
    atomicAdd(out + 2, h0.z * w);
    atomicAdd(out + 3, h0.w * w);
    atomicAdd(out + 4, h1.x * w);
    atomicAdd(out + 5, h1.y * w);
    atomicAdd(out + 6, h1.z * w);
    atomicAdd(out + 7, h1.w * w);
}

// ---------- final gather ----------
__global__ void gather_out_kernel(const float* __restrict__ h,
                                  const int* __restrict__ idx,
                                  float* __restrict__ out, int n_elems) {
    int t = blockIdx.x * blockDim.x + threadIdx.x;
    if (t >= n_elems) return;
    int g = t / HID, j = t % HID;
    out[t] = h[(size_t)idx[g] * HID + j];
}

extern "C" void kernel_launch(void* const* d_in, const int* in_sizes, int n_in,
                              void* d_out, int out_size, void* d_ws, size_t ws_size,
                              hipStream_t stream) {
    const float* x          = (const float*)d_in[0];
    const int*   edge_index = (const int*)  d_in[1];
    const int*   indices    = (const int*)  d_in[2];
    const float* Ws[4] = {(const float*)d_in[3], (const float*)d_in[5],
                          (const float*)d_in[7], (const float*)d_in[9]};
    const float* bs[4] = {(const float*)d_in[4], (const float*)d_in[6],
                          (const float*)d_in[8], (const float*)d_in[10]};

    const int n_nodes  = in_sizes[0];
    const int n_edges  = in_sizes[1] / 2;
    const int n_graphs = in_sizes[2];

    const int* srcp = edge_index;
    const int* dstp = edge_index + n_edges;

    // workspace: dis | h_lin | B0 | B1   (= 25*N floats = 100 MB at N=1M)
    float* ws    = (float*)d_ws;
    float* dis   = ws;
    float* h_lin = ws + (size_t)n_nodes;
    float* B0    = h_lin + (size_t)n_nodes * HID;
    float* B1    = B0    + (size_t)n_nodes * HID;

    const int T = 256;
    int nodeBlk = (n_nodes + T - 1) / T;
    int edgeBlk = (n_edges + T - 1) / T;

    // degree -> deg_inv_sqrt (in place in `dis`)
    zero_f32_kernel<<<nodeBlk, T, 0, stream>>>(dis, n_nodes);
    deg_accum_kernel<<<edgeBlk, T, 0, stream>>>(dstp, n_edges, dis);
    deg_to_dis_kernel<<<nodeBlk, T, 0, stream>>>(dis, n_nodes);

    // WMMA grid: one 16-node tile per wave, 8 waves per 256-thread block
    int tiles = (n_nodes + 15) / 16;
    int mmBlk = (tiles + 7) / 8;

    const float* hin = x;
    float* cur = B0;
    float* nxt = B1;
    for (int layer = 0; layer < 4; ++layer) {
        if (layer == 0) {
            layer0_transform_kernel<<<nodeBlk, T, 0, stream>>>(
                x, Ws[0], bs[0], dis, h_lin, cur, n_nodes);
        } else {
            gcn_transform_wmma8_kernel<<<mmBlk, T, 0, stream>>>(
                hin, Ws[layer], bs[layer], dis, h_lin, cur, n_nodes);
        }
        gcn_scatter_kernel<<<edgeBlk, T, 0, stream>>>(
            srcp, dstp, dis, h_lin, cur, n_edges);
        hin = cur;
        float* tmp = cur; cur = nxt; nxt = tmp;
    }

    int outElems = n_graphs * HID;
    gather_out_kernel<<<(outElems + T - 1) / T, T, 0, stream>>>(
        hin, indices, (float*)d_out, outElems);
}